// SisoMmsePicDetector_8924942041973
// MI455X (gfx1250) — compile-verified
//
#include <hip/hip_runtime.h>
#include <hip/hip_bf16.h>
#include <math.h>

typedef float v2f __attribute__((ext_vector_type(2)));
typedef float v8f __attribute__((ext_vector_type(8)));

#define Bn 16
#define Tn 12
#define Fn 1024
#define Mn 8
#define Sn 4
#define Kn 4
#define NPROB (Bn*Tn*Fn)                 /* 196608 independent subcarrier problems */
#define LLRE_SZ (Bn*Sn*Tn*Fn*Kn)         /* 3145728 floats */
#define YMF_OFF LLRE_SZ
#define YMF_SZ (NPROB*Sn*2)              /* complex64 -> 2 floats */
#define G_OFF (YMF_OFF + YMF_SZ)
#define EPSF 1e-4f
#define INV_SQRT10 0.316227766016838f

// DPP16 row_xmask: lane <- lane_in_row ^ MASK, pure VALU (no LDS pipe).
// All reduction domains here are within a 16-lane DPP row (masks <= 8).
template <int MASK>
__device__ __forceinline__ float dpp_xor(float x) {
  return __int_as_float(__builtin_amdgcn_update_dpp(
      0, __float_as_int(x), 0x160 | MASK, 0xF, 0xF, true));
}
__device__ __forceinline__ float red8_sum(float x) {
  x += dpp_xor<1>(x); x += dpp_xor<2>(x); x += dpp_xor<4>(x);
  return x;
}
__device__ __forceinline__ float red16_sum(float x) {
  x += dpp_xor<1>(x); x += dpp_xor<2>(x); x += dpp_xor<4>(x); x += dpp_xor<8>(x);
  return x;
}
__device__ __forceinline__ float red16_max(float x) {
  x = fmaxf(x, dpp_xor<1>(x)); x = fmaxf(x, dpp_xor<2>(x));
  x = fmaxf(x, dpp_xor<4>(x)); x = fmaxf(x, dpp_xor<8>(x));
  return x;
}

__global__ __launch_bounds__(256) void mmse_pic_kernel(
    const float* __restrict__ y_r, const float* __restrict__ y_i,
    const float* __restrict__ h_r, const float* __restrict__ h_i,
    const float* __restrict__ no,  const float* __restrict__ llr_a,
    float* __restrict__ out)
{
  __shared__ float sG[32][4][4][2];  // Gram matrix per problem (complex)
  __shared__ float sY[32][4][2];     // matched filter output
  __shared__ float sS[32][4][2];     // soft symbols s_hat
  __shared__ float sE[32][4];        // err_var
  __shared__ float sX[32][4][2];     // x_hat
  __shared__ float sR[32][4];        // rho

  const int tid   = threadIdx.x;
  const int Pbase = blockIdx.x * 32;

  // ---------------- Phase 1: matched filter + Gram (8 lanes / problem) ----
  {
    const int m  = tid & 7;       // antenna
    const int pl = tid >> 3;      // local problem 0..31
    const int P  = Pbase + pl;
    __builtin_prefetch(h_r + (size_t)(P + 32) * (Mn*Sn), 0, 1);
    __builtin_prefetch(h_i + (size_t)(P + 32) * (Mn*Sn), 0, 1);
    const float4 hr4 = *(const float4*)(h_r + (size_t)P * (Mn*Sn) + m*Sn);
    const float4 hi4 = *(const float4*)(h_i + (size_t)P * (Mn*Sn) + m*Sn);
    const float yr = y_r[(size_t)P*Mn + m];
    const float yi = y_i[(size_t)P*Mn + m];
    const float hrv[4] = {hr4.x, hr4.y, hr4.z, hr4.w};
    const float hiv[4] = {hi4.x, hi4.y, hi4.z, hi4.w};
    float ymr[4], ymi[4], gr[4][4], gi[4][4];
#pragma unroll
    for (int i = 0; i < 4; ++i) {
      ymr[i] = red8_sum(hrv[i]*yr + hiv[i]*yi);     // conj(h)*y
      ymi[i] = red8_sum(hrv[i]*yi - hiv[i]*yr);
      // Hermitian: reduce upper triangle only, G[j][i] = conj(G[i][j])
#pragma unroll
      for (int j = i; j < 4; ++j) {
        gr[i][j] = red8_sum(hrv[i]*hrv[j] + hiv[i]*hiv[j]);
        gi[i][j] = red8_sum(hrv[i]*hiv[j] - hiv[i]*hrv[j]);
      }
    }
    if (m == 0) {
      float* go = out + G_OFF   + (size_t)P * 32;
      float* yo = out + YMF_OFF + (size_t)P * 8;
#pragma unroll
      for (int i = 0; i < 4; ++i) {
        sY[pl][i][0] = ymr[i]; sY[pl][i][1] = ymi[i];
        yo[i*2] = ymr[i]; yo[i*2+1] = ymi[i];
#pragma unroll
        for (int j = i; j < 4; ++j) {
          sG[pl][i][j][0] = gr[i][j]; sG[pl][i][j][1] = gi[i][j];
          sG[pl][j][i][0] = gr[i][j]; sG[pl][j][i][1] = -gi[i][j];
          go[(i*4+j)*2] = gr[i][j];   go[(i*4+j)*2+1] = gi[i][j];
          if (j != i) { go[(j*4+i)*2] = gr[i][j]; go[(j*4+i)*2+1] = -gi[i][j]; }
        }
      }
    }
  }
  __syncthreads();

  // ---------------- Phase 2: priors -> log_pc via WMMA f32 16x16x4 -------
  const int w    = tid >> 5;       // wave id: handles problems 4w..4w+3
  const int lane = tid & 31;
  const int half = lane >> 4;
  const int cpt  = lane & 15;      // constellation point owned by this lane in D
  v8f d = {};                      // log_pc tile: 16 rows x 16 points
  float prc, pic, ap2c;
  {
    // A-matrix layout (f32 16x4): lanes 0-15 row M=lane, K=0,1; lanes 16-31 row M=lane-16, K=2,3
    const int row = lane & 15;         // (problem,stream) row
    const int k0  = half * 2;          // bit index base
    const int pl  = 4*w + (row >> 2);
    const int s   = row & 3;
    const int P   = Pbase + pl;
    const int b   = P / (Tn*Fn);
    const int t   = (P / Fn) % Tn;
    const int f   = P & (Fn-1);
    const size_t laIdx = ((((size_t)b*Sn + s)*Tn + t)*Fn + f)*Kn + k0;
    const float la0 = llr_a[laIdx];
    const float la1 = llr_a[laIdx + 1];
    // logp1 = -softplus(-x) = min(x,0)-log1p(e^-|x|) ; logp0 = -softplus(x)
    const float c0 = log1pf(expf(-fabsf(la0)));
    const float c1 = log1pf(expf(-fabsf(la1)));
    v2f A1, A0, B1, B0;
    A1.x =  fminf(la0, 0.f) - c0;  A1.y =  fminf(la1, 0.f) - c1;
    A0.x = -fmaxf(la0, 0.f) - c0;  A0.y = -fmaxf(la1, 0.f) - c1;
    // B-matrix (4x16): B[k][c] = bit k of point c (MSB-first per np.binary_repr)
    const float bb0 = (float)((cpt >> (3 - k0)) & 1);
    const float bb1 = (float)((cpt >> (2 - k0)) & 1);
    B1.x = bb0;        B1.y = bb1;
    B0.x = 1.f - bb0;  B0.y = 1.f - bb1;
    d = __builtin_amdgcn_wmma_f32_16x16x4_f32(false, A1, false, B1, (short)0, d, false, false);
    d = __builtin_amdgcn_wmma_f32_16x16x4_f32(false, A0, false, B0, (short)0, d, false, false);

    // Gray-mapped 16-QAM point for this lane: re from bits(0,2), im from bits(1,3)
    const int cb0 = (cpt >> 3) & 1, cb1 = (cpt >> 2) & 1;
    const int cb2 = (cpt >> 1) & 1, cb3 = cpt & 1;
    prc  = (float)((1 - 2*cb0) * (1 + 2*cb2)) * INV_SQRT10;
    pic  = (float)((1 - 2*cb1) * (1 + 2*cb3)) * INV_SQRT10;
    ap2c = prc*prc + pic*pic;

    // soft-symbol statistics: reduce over the 16 constellation lanes
#pragma unroll
    for (int v = 0; v < 8; ++v) {
      const float p  = expf(d[v]);
      const float sr = red16_sum(p * prc);
      const float si = red16_sum(p * pic);
      const float e2 = red16_sum(p * ap2c);
      if (cpt == 0) {
        const int rowa = half*8 + v;          // D-layout: M = v + 8*half
        const int pla  = 4*w + (rowa >> 2);
        const int sa   = rowa & 3;
        sS[pla][sa][0] = sr;
        sS[pla][sa][1] = si;
        sE[pla][sa]    = e2 - (sr*sr + si*si);
      }
    }
  }
  __syncthreads();

  // ---------------- Phase 3: 4x4 complex inverse + MMSE stats ------------
  if (tid < 32) {
    const int pl = tid;
    const int P  = Pbase + pl;
    const int b  = P / (Tn*Fn);
    const float nof = no[b];
    float ar[4][4], ai[4][4], br[4][4], bi[4][4];
#pragma unroll
    for (int i = 0; i < 4; ++i)
#pragma unroll
      for (int j = 0; j < 4; ++j) {
        ar[i][j] = sG[pl][i][j][0]*sE[pl][j] + ((i==j) ? nof : 0.f);
        ai[i][j] = sG[pl][i][j][1]*sE[pl][j];
        br[i][j] = (i==j) ? 1.f : 0.f;
        bi[i][j] = 0.f;
      }
    // Gauss-Jordan, no pivoting (A is diagonally loaded by no > 0)
#pragma unroll
    for (int c = 0; c < 4; ++c) {
      const float pr = ar[c][c], pq = ai[c][c];
      const float den = 1.f / (pr*pr + pq*pq + 1e-30f);
      const float ivr = pr*den, ivi = -pq*den;
#pragma unroll
      for (int j = 0; j < 4; ++j) {
        float tr = ar[c][j], ti = ai[c][j];
        ar[c][j] = tr*ivr - ti*ivi; ai[c][j] = tr*ivi + ti*ivr;
        tr = br[c][j]; ti = bi[c][j];
        br[c][j] = tr*ivr - ti*ivi; bi[c][j] = tr*ivi + ti*ivr;
      }
#pragma unroll
      for (int r = 0; r < 4; ++r) {
        if (r == c) continue;
        const float fr = ar[r][c], fi = ai[r][c];
#pragma unroll
        for (int j = 0; j < 4; ++j) {
          const float acr = ar[c][j], aci = ai[c][j];
          const float bcr = br[c][j], bci = bi[c][j];
          ar[r][j] -= fr*acr - fi*aci;  ai[r][j] -= fr*aci + fi*acr;
          br[r][j] -= fr*bcr - fi*bci;  bi[r][j] -= fr*bci + fi*bcr;
        }
      }
    }
    float mu[4], ybr[4], ybi[4];
#pragma unroll
    for (int i = 0; i < 4; ++i) {
      float acc = 0.f;
#pragma unroll
      for (int j = 0; j < 4; ++j)
        acc += br[i][j]*sG[pl][j][i][0] - bi[i][j]*sG[pl][j][i][1];
      mu[i] = acc;
    }
#pragma unroll
    for (int i = 0; i < 4; ++i) {
      float accr = sY[pl][i][0], acci = sY[pl][i][1];
#pragma unroll
      for (int j = 0; j < 4; ++j) {
        const float gr2 = sG[pl][i][j][0], gi2 = sG[pl][i][j][1];
        const float sr  = sS[pl][j][0],   si  = sS[pl][j][1];
        accr -= gr2*sr - gi2*si;
        acci -= gr2*si + gi2*sr;
      }
      ybr[i] = accr; ybi[i] = acci;
    }
#pragma unroll
    for (int i = 0; i < 4; ++i) {
      float accr = mu[i]*sS[pl][i][0], acci = mu[i]*sS[pl][i][1];
#pragma unroll
      for (int j = 0; j < 4; ++j) {
        accr += br[i][j]*ybr[j] - bi[i][j]*ybi[j];
        acci += br[i][j]*ybi[j] + bi[i][j]*ybr[j];
      }
      const float m = fmaxf(mu[i], EPSF);
      sX[pl][i][0] = accr / m;
      sX[pl][i][1] = acci / m;
      sR[pl][i]    = mu[i] / fmaxf(1.f - sE[pl][i]*mu[i], EPSF);
    }
  }
  __syncthreads();

  // ---------------- Phase 4: per-bit logsumexp -> extrinsic LLRs ---------
  // llr_k = log(sum_{bit_k=1} p) - log(sum p - sum_{bit_k=1} p); the shared
  // row-max cancels in the difference, so one max + one exp per row suffice.
  {
#pragma unroll
    for (int v = 0; v < 8; ++v) {
      const int row = half*8 + v;
      const int pl  = 4*w + (row >> 2);
      const int s   = row & 3;
      const float xr  = sX[pl][s][0], xi = sX[pl][s][1];
      const float rho = sR[pl][s];
      const float dr = xr - prc, di = xi - pic;
      const float e  = -rho*(dr*dr + di*di) + d[v];
      const float m  = red16_max(e);
      const float p  = expf(e - m);
      const float stot = red16_sum(p);
      float s1[4];
#pragma unroll
      for (int k = 0; k < 4; ++k) {
        const int bit = (cpt >> (3 - k)) & 1;
        s1[k] = red16_sum(bit ? p : 0.f);
      }
#pragma unroll
      for (int k = 0; k < 4; ++k) {
        if (cpt == k) {
          const float llr_d = logf(s1[k]) - logf(stot - s1[k]);
          const int P = Pbase + pl;
          const int b = P / (Tn*Fn);
          const int t = (P / Fn) % Tn;
          const int f = P & (Fn-1);
          const size_t oIdx = ((((size_t)b*Sn + s)*Tn + t)*Fn + f)*Kn + k;
          out[oIdx] = llr_d - llr_a[oIdx];
        }
      }
    }
  }
}

extern "C" void kernel_launch(void* const* d_in, const int* in_sizes, int n_in,
                              void* d_out, int out_size, void* d_ws, size_t ws_size,
                              hipStream_t stream) {
  const float* y_r   = (const float*)d_in[0];
  const float* y_i   = (const float*)d_in[1];
  const float* h_r   = (const float*)d_in[2];
  const float* h_i   = (const float*)d_in[3];
  const float* no    = (const float*)d_in[4];
  const float* llr_a = (const float*)d_in[5];
  float* out = (float*)d_out;
  dim3 grid(NPROB / 32);
  dim3 block(256);
  mmse_pic_kernel<<<grid, block, 0, stream>>>(y_r, y_i, h_r, h_i, no, llr_a, out);
}